// SegmentedParallelLorentzMovingWindow_41790031790538
// MI455X (gfx1250) — compile-verified
//
#include <hip/hip_runtime.h>
#include <math.h>

// ---------------------------------------------------------------------------
// SegmentedParallelLorentzMovingWindow — fully fused MI455X (gfx1250) kernel
//   encode (4x GEMM [262144,24]x[24,32] via V_WMMA_F32_16X16X4_F32)
//   -> to_hyperbolic (expmap0 + projx)  -> Lorentz Frechet-mean fusion (10 it)
// Memory-bound: ~270MB traffic -> single pass, f32 end-to-end.
// ---------------------------------------------------------------------------

typedef float v2f __attribute__((ext_vector_type(2)));
typedef float v8f __attribute__((ext_vector_type(8)));

#define B_DIM   2048
#define L_DIM   3072
#define SEG     24
#define NSEG    (L_DIM / SEG)            // 128
#define DD      32
#define ROWS    (B_DIM * NSEG)           // 262144
#define OUTSZ   ((size_t)ROWS * DD)      // 8388608 floats per output tensor
#define EPSF    1e-7f
#define ROWS_PER_WG 256                  // 8 waves x 32 rows
#define ZSTRIDE 36                       // 144B row stride: 16B aligned, bank-friendly

__launch_bounds__(256)
__global__ void slmw_fused(
    const float* __restrict__ xt, const float* __restrict__ xc,
    const float* __restrict__ xf, const float* __restrict__ xr,
    const float* __restrict__ Wt, const float* __restrict__ bt,
    const float* __restrict__ Wc, const float* __restrict__ bc,
    const float* __restrict__ Wf, const float* __restrict__ bf,
    const float* __restrict__ Wr, const float* __restrict__ br,
    const float* __restrict__ scale, const float* __restrict__ lw,
    float* __restrict__ out)
{
    __shared__ float zbuf[8 * 32 * ZSTRIDE];   // 36864 B

    const int lane  = threadIdx.x & 31;
    const int wave  = threadIdx.x >> 5;
    const int rowBase = blockIdx.x * ROWS_PER_WG + wave * 32;
    float* zw = zbuf + wave * 32 * ZSTRIDE;

    const float tsc = tanhf(scale[0]);

    // softmax(lorentz_weights) — 4 scalars, per-thread
    float w4[4];
    {
        float l0 = lw[0], l1 = lw[1], l2 = lw[2], l3 = lw[3];
        float mx = fmaxf(fmaxf(l0, l1), fmaxf(l2, l3));
        float e0 = expf(l0 - mx), e1 = expf(l1 - mx), e2 = expf(l2 - mx), e3 = expf(l3 - mx);
        float inv = 1.0f / (e0 + e1 + e2 + e3);
        w4[0] = e0 * inv; w4[1] = e1 * inv; w4[2] = e2 * inv; w4[3] = e3 * inv;
    }

    const float* Xs[4] = {xt, xc, xf, xr};
    const float* Ws[4] = {Wt, Wc, Wf, Wr};
    const float* Bs[4] = {bt, bc, bf, br};

    // f32 WMMA 16x16x4 operand lane mapping:
    //  A (16x4):  lanes 0-15 -> M=lane, K={k0,k0+1}; lanes 16-31 -> M=lane-16, K={k0+2,k0+3}
    //  B (4x16):  lanes 0-15 -> N=lane, K={k0,k0+1}; lanes 16-31 -> N=lane-16, K={k0+2,k0+3}
    const int nLane = lane & 15;
    const int kOff  = (lane < 16) ? 0 : 2;

    float h[4][DD];   // four hyperboloid points; this lane's row

    #pragma unroll
    for (int s = 0; s < 4; ++s) {
        // ---- B fragments: W^T (24x32) as 2 col-halves x 6 K-chunks; B[k][n]=W[n][k]
        v2f bf0[6], bf1[6];
        #pragma unroll
        for (int ck = 0; ck < 6; ++ck) {
            bf0[ck] = *(const v2f*)(Ws[s] + (nLane     ) * SEG + ck * 4 + kOff);
            bf1[ck] = *(const v2f*)(Ws[s] + (nLane + 16) * SEG + ck * 4 + kOff);
        }
        const float bias0 = Bs[s][nLane];
        const float bias1 = Bs[s][nLane + 16];

        #pragma unroll
        for (int t = 0; t < 2; ++t) {          // two 16-row tiles per wave
            v8f c0, c1;                        // C/D: lane owns one column, 8 rows
            #pragma unroll
            for (int j = 0; j < 8; ++j) { c0[j] = bias0; c1[j] = bias1; }
            #pragma unroll
            for (int ck = 0; ck < 6; ++ck) {   // K = 24 = 6 x 4
                const float* ap = Xs[s]
                    + (size_t)(rowBase + t * 16 + nLane) * SEG + ck * 4 + kOff;
                v2f a = *(const v2f*)ap;
                c0 = __builtin_amdgcn_wmma_f32_16x16x4_f32(
                        false, a, false, bf0[ck], (short)0, c0, false, false);
                c1 = __builtin_amdgcn_wmma_f32_16x16x4_f32(
                        false, a, false, bf1[ck], (short)0, c1, false, false);
            }
            // transpose D-layout (col per lane) -> row-major z tile in LDS
            const int mb = t * 16 + ((lane < 16) ? 0 : 8);
            #pragma unroll
            for (int j = 0; j < 8; ++j) {
                zw[(mb + j) * ZSTRIDE + nLane     ] = c0[j];
                zw[(mb + j) * ZSTRIDE + nLane + 16] = c1[j];
            }
        }
        __syncthreads();

        // ---- lane-per-row: u = z*tanh(scale); h = projx(safe_expmap0(u))
        float u[DD];
        {
            const float4* zr = (const float4*)(zw + lane * ZSTRIDE); // 144B stride: aligned
            #pragma unroll
            for (int q = 0; q < 8; ++q) {
                float4 v = zr[q];
                u[4*q+0] = v.x * tsc; u[4*q+1] = v.y * tsc;
                u[4*q+2] = v.z * tsc; u[4*q+3] = v.w * tsc;
            }
        }
        __syncthreads();   // zbuf reusable for next stream

        float sq = -u[0] * u[0];                       // Minkowski <u,u>
        #pragma unroll
        for (int i = 1; i < DD; ++i) sq = fmaf(u[i], u[i], sq);
        sq = fmaxf(sq, 1e-12f);
        float nrm = sqrtf(sq);
        float f   = sinhf(nrm) / fmaxf(nrm, EPSF);
        float ssp = 0.0f;
        #pragma unroll
        for (int i = 1; i < DD; ++i) {
            float hv = f * u[i]; h[s][i] = hv; ssp = fmaf(hv, hv, ssp);
        }
        h[s][0] = sqrtf(1.0f + ssp);                   // projx recomputes time

        float* op = out + (size_t)s * OUTSZ + (size_t)(rowBase + lane) * DD;
        #pragma unroll
        for (int q = 0; q < 8; ++q) {
            float4 v; v.x = h[s][4*q]; v.y = h[s][4*q+1];
            v.z = h[s][4*q+2]; v.w = h[s][4*q+3];
            ((float4*)op)[q] = v;
        }
    }

    // ---- Lorentz fusion: t0 = sum_s w_s*logmap0(h_s); mean0 = projx(expmap0(t0))
    float mean[DD];
    {
        float t0[DD];
        #pragma unroll
        for (int i = 0; i < DD; ++i) t0[i] = 0.0f;
        #pragma unroll
        for (int s = 0; s < 4; ++s) {
            float a  = fmaxf(h[s][0], 1.0f + EPSF);
            float cf = acoshf(a) / sqrtf(a * a - 1.0f);
            float wc = w4[s] * cf;
            t0[0] += wc * (h[s][0] - a);
            #pragma unroll
            for (int i = 1; i < DD; ++i) t0[i] = fmaf(wc, h[s][i], t0[i]);
        }
        float sq = -t0[0] * t0[0];
        #pragma unroll
        for (int i = 1; i < DD; ++i) sq = fmaf(t0[i], t0[i], sq);
        sq = fmaxf(sq, 1e-12f);
        float nrm = sqrtf(sq);
        float f   = sinhf(nrm) / fmaxf(nrm, EPSF);
        float ssp = 0.0f;
        #pragma unroll
        for (int i = 1; i < DD; ++i) {
            float mi = f * t0[i]; mean[i] = mi; ssp = fmaf(mi, mi, ssp);
        }
        mean[0] = sqrtf(1.0f + ssp);
    }

    // ---- 10 Frechet-mean refinement steps (global 'done' never fires pre-tol)
    for (int it = 0; it < 10; ++it) {
        float acc[DD];
        #pragma unroll
        for (int i = 0; i < DD; ++i) acc[i] = 0.0f;
        float btot = 0.0f;
        #pragma unroll
        for (int s = 0; s < 4; ++s) {
            float al = mean[0] * h[s][0];              // -<mean,h>_L
            #pragma unroll
            for (int i = 1; i < DD; ++i) al = fmaf(-mean[i], h[s][i], al);
            al = fmaxf(al, 1.0f + EPSF);
            float cf  = acoshf(al) / sqrtf(al * al - 1.0f);
            float a_s = w4[s] * cf;
            btot = fmaf(a_s, al, btot);
            #pragma unroll
            for (int i = 0; i < DD; ++i) acc[i] = fmaf(a_s, h[s][i], acc[i]);
        }
        // wv[i] = acc[i] - btot*mean[i]; u = 0.5*wv  (recompute u, no extra array)
        float sq;
        {
            float u0 = 0.5f * fmaf(-btot, mean[0], acc[0]);
            sq = -u0 * u0;
        }
        #pragma unroll
        for (int i = 1; i < DD; ++i) {
            float ui = 0.5f * fmaf(-btot, mean[i], acc[i]);
            sq = fmaf(ui, ui, sq);
        }
        sq = fmaxf(sq, 1e-12f);
        float nrm = sqrtf(sq);
        float ch  = coshf(nrm);
        float g   = 0.5f * (sinhf(nrm) / fmaxf(nrm, EPSF));  // sh * 0.5 folded
        float cm  = ch - g * btot;       // mi = cm*mean[i] + g*acc[i]
        float ssp = 0.0f;
        #pragma unroll
        for (int i = 1; i < DD; ++i) {
            float mi = fmaf(cm, mean[i], g * acc[i]);
            mean[i] = mi; ssp = fmaf(mi, mi, ssp);
        }
        mean[0] = sqrtf(1.0f + ssp);     // projx
    }

    float* op = out + (size_t)4 * OUTSZ + (size_t)(rowBase + lane) * DD;
    #pragma unroll
    for (int q = 0; q < 8; ++q) {
        float4 v; v.x = mean[4*q]; v.y = mean[4*q+1];
        v.z = mean[4*q+2]; v.w = mean[4*q+3];
        ((float4*)op)[q] = v;
    }
}

extern "C" void kernel_launch(void* const* d_in, const int* in_sizes, int n_in,
                              void* d_out, int out_size, void* d_ws, size_t ws_size,
                              hipStream_t stream) {
    (void)in_sizes; (void)n_in; (void)out_size; (void)d_ws; (void)ws_size;
    dim3 grid(ROWS / ROWS_PER_WG);   // 1024 blocks, exact cover -> EXEC all-ones
    dim3 block(256);                 // 8 waves (wave32)
    slmw_fused<<<grid, block, 0, stream>>>(
        (const float*)d_in[0],  (const float*)d_in[1],
        (const float*)d_in[2],  (const float*)d_in[3],
        (const float*)d_in[4],  (const float*)d_in[5],
        (const float*)d_in[6],  (const float*)d_in[7],
        (const float*)d_in[8],  (const float*)d_in[9],
        (const float*)d_in[10], (const float*)d_in[11],
        (const float*)d_in[12], (const float*)d_in[13],
        (float*)d_out);
}